// DiagonalAffine_30021821399532
// MI455X (gfx1250) — compile-verified
//
#include <hip/hip_runtime.h>

// out[r][c] = x[r][c] * A[c] + B[c]   (N=262144 rows, D=512 cols, fp32)
//
// Memory-bound: 1.07 GB single-touch traffic @ 23.3 TB/s => ~46 us floor.
// Strategy: wave32 b128 NT streaming loads/stores, A/B cached in registers
// per thread (fixed column assignment), 8 rows in flight per thread.

typedef __attribute__((ext_vector_type(4))) float v4f;

#define D_COLS   512
#define D4       128          // float4 columns per row
#define BLOCK    256          // 8 wave32 waves
#define ROWSUB   2            // 256 threads / 128 col-threads = 2 row phases
#define UNROLL   8            // rows per thread per iteration
#define ROWS_PER_BLOCK (ROWSUB * UNROLL)   // 16

__global__ __launch_bounds__(BLOCK) void diag_affine_kernel(
    const v4f* __restrict__ x4,
    const float* __restrict__ A,
    const float* __restrict__ B,
    v4f* __restrict__ out4,
    int n_rows)
{
    const int col4 = threadIdx.x & (D4 - 1);   // fixed 4-col slice for this thread
    const int rsub = threadIdx.x >> 7;         // 0 or 1: row phase within block

    // Scale/shift for this thread's columns: loaded ONCE, lives in VGPRs.
    const v4f a = reinterpret_cast<const v4f*>(A)[col4];
    const v4f b = reinterpret_cast<const v4f*>(B)[col4];

    const long long stride = (long long)gridDim.x * ROWS_PER_BLOCK;
    long long r = (long long)blockIdx.x * ROWS_PER_BLOCK + rsub;

    for (; r + (long long)ROWSUB * (UNROLL - 1) < n_rows; r += stride) {
        // Fast path: no per-row guards -> loads cluster into a clause,
        // 8 outstanding global_load_b128 (NT) per thread.
        v4f v[UNROLL];
        long long idx[UNROLL];
#pragma unroll
        for (int k = 0; k < UNROLL; ++k) {
            idx[k] = (r + (long long)ROWSUB * k) * D4 + col4;
            v[k] = __builtin_nontemporal_load(&x4[idx[k]]);
        }
#pragma unroll
        for (int k = 0; k < UNROLL; ++k) {
            v4f o;
            o.x = fmaf(v[k].x, a.x, b.x);
            o.y = fmaf(v[k].y, a.y, b.y);
            o.z = fmaf(v[k].z, a.z, b.z);
            o.w = fmaf(v[k].w, a.w, b.w);
            __builtin_nontemporal_store(o, &out4[idx[k]]);
        }
    }

    // Guarded tail (only runs if n_rows is not a multiple of the sweep).
    for (; r < n_rows; r += stride) {
#pragma unroll
        for (int k = 0; k < UNROLL; ++k) {
            long long rr = r + (long long)ROWSUB * k;
            if (rr < n_rows) {
                long long i = rr * D4 + col4;
                v4f v = __builtin_nontemporal_load(&x4[i]);
                v4f o;
                o.x = fmaf(v.x, a.x, b.x);
                o.y = fmaf(v.y, a.y, b.y);
                o.z = fmaf(v.z, a.z, b.z);
                o.w = fmaf(v.w, a.w, b.w);
                __builtin_nontemporal_store(o, &out4[i]);
            }
        }
    }
}

extern "C" void kernel_launch(void* const* d_in, const int* in_sizes, int n_in,
                              void* d_out, int out_size, void* d_ws, size_t ws_size,
                              hipStream_t stream)
{
    const float* x = (const float*)d_in[0];   // (N, 512) fp32
    const float* A = (const float*)d_in[1];   // (512,)   fp32
    const float* B = (const float*)d_in[2];   // (512,)   fp32
    float* out = (float*)d_out;               // (N, 512) fp32

    const long long total = (long long)in_sizes[0];
    const int n_rows = (int)(total / D_COLS); // 262144 for the reference shape

    long long grid_ll = (n_rows + ROWS_PER_BLOCK - 1) / ROWS_PER_BLOCK;
    if (grid_ll > 8192) grid_ll = 8192;       // grid-stride: 16 rows/block/iter
    if (grid_ll < 1) grid_ll = 1;
    dim3 grid((unsigned)grid_ll), block(BLOCK);

    diag_affine_kernel<<<grid, block, 0, stream>>>(
        (const v4f*)x, A, B, (v4f*)out, n_rows);
}